// LearnableLaplacianBasis_3865470567001
// MI455X (gfx1250) — compile-verified
//
#include <hip/hip_runtime.h>
#include <math.h>

typedef __attribute__((ext_vector_type(2))) float v2f;
typedef __attribute__((ext_vector_type(4))) float v4f;
typedef __attribute__((ext_vector_type(8))) float v8f;
typedef __attribute__((ext_vector_type(4))) int   v4i;

#define BB 8
#define TT 8192
#define DD 128
#define KK 64
#define NC 64     // chunks along T
#define CL 128    // chunk length (NC*CL == TT)

#define WR 132    // padded W row (floats): 528B, 16B aligned, bank stride 4
#define XR 20     // padded x-tile row (floats): 80B, 16B aligned, bank stride 20
#define KCH 16    // K columns staged per async chunk
#define NKCH (DD / KCH)

// ---- async global->LDS helpers (guarded; sync fallback keeps compile safe) --
#if __has_builtin(__builtin_amdgcn_global_load_async_to_lds_b128) && \
    __has_builtin(__builtin_amdgcn_s_wait_asynccnt)
#define USE_ASYNC_LDS 1
#else
#define USE_ASYNC_LDS 0
#endif

__device__ __forceinline__ void copy16_g2l(const float* g, float* l) {
#if USE_ASYNC_LDS
    __builtin_amdgcn_global_load_async_to_lds_b128(
        (__attribute__((address_space(1))) v4i*)(uintptr_t)g,
        (__attribute__((address_space(3))) v4i*)(uintptr_t)l,
        0, 0);
#else
    *(v4f*)l = *(const v4f*)g;
#endif
}

__device__ __forceinline__ void wait_async_lds() {
#if USE_ASYNC_LDS
    __builtin_amdgcn_s_wait_asynccnt(0);
#endif
}

__device__ __forceinline__ float softplus_f(float x) {
    return logf(1.0f + __expf(x));
}

// ---------------------------------------------------------------------------
// Kernel 1: power iteration for sigma_max(W), plus per-mode constants.
// misc[0] = 1/sigma ; misc[64+j] = alpha0[j] ; misc[128+j] = omega0[j]
// ---------------------------------------------------------------------------
__global__ void __launch_bounds__(128)
k_spectral(const float* __restrict__ W, const float* __restrict__ s_real,
           const float* __restrict__ s_imag, const float* __restrict__ tau_raw,
           float* __restrict__ misc)
{
    __shared__ float v[DD];
    __shared__ float uu[KK];
    __shared__ float red[DD];
    __shared__ float s_tau;
    const int tid = threadIdx.x;           // 128 threads

    v[tid] = 1.0f + 0.01f * (float)tid;    // symmetry-broken init
    __syncthreads();

    float norm2 = 0.0f;
    for (int it = 0; it < 40; ++it) {
        if (tid < KK) {
            float acc = 0.0f;
            for (int d = 0; d < DD; ++d) acc += W[tid * DD + d] * v[d];
            uu[tid] = acc;
        }
        __syncthreads();
        float w = 0.0f;
        for (int j = 0; j < KK; ++j) w += W[j * DD + tid] * uu[j];
        red[tid] = w * w;
        __syncthreads();
        for (int s = DD / 2; s > 0; s >>= 1) {
            if (tid < s) red[tid] += red[tid + s];
            __syncthreads();
        }
        norm2 = red[0];                    // = lambda^2 at convergence
        v[tid] = w * rsqrtf(norm2);
        __syncthreads();
    }
    if (tid == 0) {
        misc[0] = rsqrtf(sqrtf(norm2));    // norm2^(-1/4) = 1/sigma
        s_tau = softplus_f(tau_raw[0]) + 1e-3f;
    }
    __syncthreads();
    if (tid < KK) {
        const float tau = s_tau;
        misc[64 + tid]  = (softplus_f(s_real[tid]) + 1e-6f) * tau; // alpha0
        misc[128 + tid] = s_imag[tid] * tau;                        // omega0
    }
}

// ---------------------------------------------------------------------------
// Kernel 2: u = (x @ W^T) * inv_sigma + b via V_WMMA_F32_16X16X4_F32.
// Block = 256 threads (8 waves); each wave owns a 16x64 output strip.
// Raw W staged once in LDS (padded rows, no bank conflicts) and x staged in
// double-buffered K-slices -- both through the async global->LDS engine.
// ---------------------------------------------------------------------------
__global__ void __launch_bounds__(256)
k_gemm(const float* __restrict__ x, const float* __restrict__ W,
       const float* __restrict__ bias, const float* __restrict__ misc,
       float* __restrict__ u_ws)
{
    __shared__ __align__(16) float wl[KK * WR];        // 33.0 KB
    __shared__ __align__(16) float xs[2][128 * XR];    // 2 x 10.0 KB
    __shared__ float bl[KK];
    const int tid = threadIdx.x;
    const float inv_sigma = misc[0];
    const int mblk = blockIdx.x * 128;

    // stage raw W: 64 rows x 32 16B-chunks = 2048 transfers, 8 per thread
    #pragma unroll
    for (int p = 0; p < 8; ++p) {
        const int q = tid + 256 * p;
        const int row = q >> 5, ch = q & 31;
        copy16_g2l(W + row * DD + ch * 4, &wl[row * WR + ch * 4]);
    }
    // stage x K-chunk 0: 128 rows x 4 16B-chunks = 512 transfers, 2 per thread
    #pragma unroll
    for (int p = 0; p < 2; ++p) {
        const int q = tid + 256 * p;
        const int row = q >> 2, c4 = q & 3;
        copy16_g2l(x + (size_t)(mblk + row) * DD + c4 * 4,
                   &xs[0][row * XR + c4 * 4]);
    }
    if (tid < KK) bl[tid] = bias[tid];

    const int wave = tid >> 5;
    const int lane = tid & 31;
    const int half = (lane >> 4) & 1;      // lanes 16-31 carry K+2 / M+8
    const int lr   = lane & 15;
    const int arow = wave * 16 + lr;       // this lane's A row within the block

    v8f acc0 = {}, acc1 = {}, acc2 = {}, acc3 = {};

    for (int c = 0; c < NKCH; ++c) {
        wait_async_lds();
        __syncthreads();                    // chunk c resident; prev buf free
        if (c + 1 < NKCH) {
            const int nb = (c + 1) & 1;
            #pragma unroll
            for (int p = 0; p < 2; ++p) {
                const int q = tid + 256 * p;
                const int row = q >> 2, c4 = q & 3;
                copy16_g2l(x + (size_t)(mblk + row) * DD + (c + 1) * KCH + c4 * 4,
                           &xs[nb][row * XR + c4 * 4]);
            }
        }
        const float* xb = xs[c & 1];
        #pragma unroll
        for (int kl = 0; kl < KCH; kl += 4) {
            const int kg = c * KCH + kl;
            // A tile 16x4: VGPR0/1 = K {0,1} (lanes 0-15), K {2,3} (lanes 16-31)
            v2f a  = *(const v2f*)(&xb[arow * XR + kl + 2 * half]);
            // B tile 4x16: B[K][N] = W[N][K]
            v2f b0 = *(const v2f*)(&wl[(0 * 16 + lr) * WR + kg + 2 * half]);
            v2f b1 = *(const v2f*)(&wl[(1 * 16 + lr) * WR + kg + 2 * half]);
            v2f b2 = *(const v2f*)(&wl[(2 * 16 + lr) * WR + kg + 2 * half]);
            v2f b3 = *(const v2f*)(&wl[(3 * 16 + lr) * WR + kg + 2 * half]);
            acc0 = __builtin_amdgcn_wmma_f32_16x16x4_f32(false, a, false, b0, (short)0, acc0, false, false);
            acc1 = __builtin_amdgcn_wmma_f32_16x16x4_f32(false, a, false, b1, (short)0, acc1, false, false);
            acc2 = __builtin_amdgcn_wmma_f32_16x16x4_f32(false, a, false, b2, (short)0, acc2, false, false);
            acc3 = __builtin_amdgcn_wmma_f32_16x16x4_f32(false, a, false, b3, (short)0, acc3, false, false);
        }
    }

    // C/D layout: VGPR r -> row m0 + r + 8*half, col = nbase + lr
    const int m0 = mblk + wave * 16;
    #pragma unroll
    for (int r = 0; r < 8; ++r) {
        const int m = m0 + r + 8 * half;
        float* up = u_ws + (size_t)m * KK;
        up[0 * 16 + lr] = acc0[r] * inv_sigma + bl[0 * 16 + lr];
        up[1 * 16 + lr] = acc1[r] * inv_sigma + bl[1 * 16 + lr];
        up[2 * 16 + lr] = acc2[r] * inv_sigma + bl[2 * 16 + lr];
        up[3 * 16 + lr] = acc3[r] * inv_sigma + bl[3 * 16 + lr];
    }
}

// ---------------------------------------------------------------------------
// Kernel 3a: per-chunk local scan (zero init) -> chunk carry (A_prod, z_end).
// 32768 chains: gid -> j (fast, coalesced), chunk, batch. Carries only.
// ---------------------------------------------------------------------------
__global__ void __launch_bounds__(256)
k_scan1(const float* __restrict__ alpha_mod, const float* __restrict__ omega_mod,
        const float* __restrict__ tau_mod, const float* __restrict__ dt,
        const float* __restrict__ misc, const float* __restrict__ u_ws,
        float* __restrict__ carry)
{
    const int gid = blockIdx.x * blockDim.x + threadIdx.x;   // 32768
    const int j = gid & (KK - 1);
    const int c = (gid >> 6) & (NC - 1);
    const int b = gid >> 12;
    const float a0 = misc[64 + j];
    const float o0 = misc[128 + j];

    float zr = 0.f, zi = 0.f, Ar = 1.f, Ai = 0.f;
    const int t0 = c * CL;
    for (int tt = 0; tt < CL; ++tt) {
        const int bt = b * TT + t0 + tt;
        const float am  = alpha_mod[(size_t)bt * KK + j];
        const float om  = omega_mod[(size_t)bt * KK + j];
        const float sc  = __expf(tau_mod[bt]);
        const float dtv = dt[bt];
        const float rho = __expf(-(a0 * __expf(am) * sc) * dtv);
        float st, ct;
        __sincosf((o0 * __expf(om) * sc) * dtv, &st, &ct);
        const float ar = rho * ct, ai = rho * st;
        const float uv = u_ws[(size_t)bt * KK + j];
        const float nzr = ar * zr - ai * zi + uv;
        const float nzi = ar * zi + ai * zr;
        zr = nzr; zi = nzi;
        const float nAr = ar * Ar - ai * Ai;
        const float nAi = ar * Ai + ai * Ar;
        Ar = nAr; Ai = nAi;
    }
    float* cp = carry + (size_t)(((b * KK) + j) * NC + c) * 4;
    cp[0] = Ar; cp[1] = Ai; cp[2] = zr; cp[3] = zi;
}

// ---------------------------------------------------------------------------
// Kernel 3b: sequential combine across chunks (512 independent chains).
// zstart[chain][c] = true state entering chunk c.
// ---------------------------------------------------------------------------
__global__ void __launch_bounds__(512)
k_scan2(const float* __restrict__ carry, float* __restrict__ zstart)
{
    const int tid = threadIdx.x;           // 512 = B*KK
    const int base = tid * NC;
    float Zr = 0.f, Zi = 0.f;
    for (int c = 0; c < NC; ++c) {
        zstart[(size_t)(base + c) * 2]     = Zr;
        zstart[(size_t)(base + c) * 2 + 1] = Zi;
        const float* cp = carry + (size_t)(base + c) * 4;
        const float Ar = cp[0], Ai = cp[1], br = cp[2], bi = cp[3];
        const float nZr = Ar * Zr - Ai * Zi + br;
        const float nZi = Ar * Zi + Ai * Zr + bi;
        Zr = nZr; Zi = nZi;
    }
}

// ---------------------------------------------------------------------------
// Kernel 3c: final pass. Re-run the recurrence seeded with the true chunk
// entry state and write out exactly once (no read-modify-write of out).
// ---------------------------------------------------------------------------
__global__ void __launch_bounds__(256)
k_scan3(const float* __restrict__ alpha_mod, const float* __restrict__ omega_mod,
        const float* __restrict__ tau_mod, const float* __restrict__ dt,
        const float* __restrict__ misc, const float* __restrict__ u_ws,
        const float* __restrict__ zstart, float* __restrict__ out)
{
    const int gid = blockIdx.x * blockDim.x + threadIdx.x;
    const int j = gid & (KK - 1);
    const int c = (gid >> 6) & (NC - 1);
    const int b = gid >> 12;

    const size_t zb = (size_t)(((b * KK) + j) * NC + c) * 2;
    float zr = zstart[zb], zi = zstart[zb + 1];

    const float a0 = misc[64 + j];
    const float o0 = misc[128 + j];
    const int t0 = c * CL;
    for (int tt = 0; tt < CL; ++tt) {
        const int bt = b * TT + t0 + tt;
        const float am  = alpha_mod[(size_t)bt * KK + j];
        const float om  = omega_mod[(size_t)bt * KK + j];
        const float sc  = __expf(tau_mod[bt]);
        const float dtv = dt[bt];
        const float rho = __expf(-(a0 * __expf(am) * sc) * dtv);
        float st, ct;
        __sincosf((o0 * __expf(om) * sc) * dtv, &st, &ct);
        const float ar = rho * ct, ai = rho * st;
        const float uv = u_ws[(size_t)bt * KK + j];
        const float nzr = ar * zr - ai * zi + uv;
        const float nzi = ar * zi + ai * zr;
        zr = nzr; zi = nzi;
        const size_t ob = (size_t)bt * (2 * KK);
        out[ob + j]      = zr;
        out[ob + KK + j] = zi;
    }
}

// ---------------------------------------------------------------------------
extern "C" void kernel_launch(void* const* d_in, const int* in_sizes, int n_in,
                              void* d_out, int out_size, void* d_ws, size_t ws_size,
                              hipStream_t stream)
{
    const float* x         = (const float*)d_in[0];
    const float* dt        = (const float*)d_in[1];
    const float* alpha_mod = (const float*)d_in[2];
    const float* omega_mod = (const float*)d_in[3];
    const float* tau_mod   = (const float*)d_in[4];
    const float* s_real    = (const float*)d_in[5];
    const float* s_imag    = (const float*)d_in[6];
    const float* tau_raw   = (const float*)d_in[7];
    const float* W         = (const float*)d_in[8];
    const float* bias      = (const float*)d_in[9];
    float* out = (float*)d_out;

    // workspace layout
    const size_t MISC_BYTES  = 2048;                          // 512 floats
    const size_t U_BYTES     = (size_t)BB * TT * KK * 4;      // 16 MB
    const size_t CARRY_BYTES = (size_t)BB * KK * NC * 4 * 4;  // 512 KB
    float* misc   = (float*)d_ws;
    float* u_ws   = (float*)((char*)d_ws + MISC_BYTES);
    float* carry  = (float*)((char*)d_ws + MISC_BYTES + U_BYTES);
    float* zstart = (float*)((char*)d_ws + MISC_BYTES + U_BYTES + CARRY_BYTES);
    (void)in_sizes; (void)n_in; (void)out_size; (void)ws_size;

    k_spectral<<<1, 128, 0, stream>>>(W, s_real, s_imag, tau_raw, misc);
    k_gemm<<<(BB * TT) / 128, 256, 0, stream>>>(x, W, bias, misc, u_ws);
    k_scan1<<<(BB * KK * NC) / 256, 256, 0, stream>>>(alpha_mod, omega_mod, tau_mod,
                                                      dt, misc, u_ws, carry);
    k_scan2<<<1, 512, 0, stream>>>(carry, zstart);
    k_scan3<<<(BB * KK * NC) / 256, 256, 0, stream>>>(alpha_mod, omega_mod, tau_mod,
                                                      dt, misc, u_ws, zstart, out);
}